// SpanRepresentation_84911503442048
// MI455X (gfx1250) — compile-verified
//
#include <hip/hip_runtime.h>
#include <stdint.h>

// SpanRepresentation: out[b,s,:] = concat(x[b,start[s],:], x[b,end[s]-1,:], wemb[bucket[s],:])
// B=8, L=512, D=768, widths 1..8 -> S=4068, row = 2*768+20 = 1556 f32.
// Pure data movement, ~202.6 MB written -> HBM-write bound (~8.7us @ 23.3 TB/s).
// Strategy: one wave32 block per (b, t) token; stage x[b,t,:] (3072B) in LDS via
// global_load_async_to_lds_b128 (NT), then fan out to all <=16 span segments that
// use this token via global_store_async_from_lds_b128 (NT). The async IOFFSET is
// added to BOTH the LDS and global addresses (CDNA5 ISA), so the
// six 512B chunks of a row share one address register pair and differ only in the
// immediate offset -- no per-chunk 64-bit VALU adds.

#define SPAN_W 8
#define DMODEL 768
#define WDIM 20
#define ROW_F (2 * DMODEL + WDIM) /* 1556 */

#define ASYNC_LD(OFF)                                                          \
  asm volatile("global_load_async_to_lds_b128 %0, %1, off offset:" #OFF        \
               " th:TH_LOAD_NT"                                                \
               :                                                               \
               : "v"(lds), "v"(gsrc)                                           \
               : "memory")

#define ASYNC_ST(OFF)                                                          \
  asm volatile("global_store_async_from_lds_b128 %0, %1, off offset:" #OFF     \
               " th:TH_STORE_NT"                                               \
               :                                                               \
               : "v"(gdst), "v"(lds)                                           \
               : "memory")

__device__ __forceinline__ void async_row_load(uint32_t lds, uint64_t gsrc) {
  ASYNC_LD(0);
  ASYNC_LD(512);
  ASYNC_LD(1024);
  ASYNC_LD(1536);
  ASYNC_LD(2048);
  ASYNC_LD(2560);
}

__device__ __forceinline__ void async_row_store(uint64_t gdst, uint32_t lds) {
  ASYNC_ST(0);
  ASYNC_ST(512);
  ASYNC_ST(1024);
  ASYNC_ST(1536);
  ASYNC_ST(2048);
  ASYNC_ST(2560);
}

__global__ __launch_bounds__(32) void SpanRepresentation_kernel(
    const float* __restrict__ x,    // (B, L, D)
    const float* __restrict__ swe,  // (14, WDIM)
    float* __restrict__ out,        // (B, S, ROW_F)
    int L, int S) {
  __shared__ __align__(16) float row[DMODEL];

  const int t = blockIdx.x;
  const int b = blockIdx.y;
  const int lane = threadIdx.x;

  // LDS byte address of this lane's 16B slot (as3 offset == low 32 bits of flat ptr).
  const uint32_t lds_lane = (uint32_t)(uintptr_t)(&row[0]) + (uint32_t)lane * 16u;

  // Stage x[b, t, :] into LDS (6 async b128 per wave, shared addr regs + imm offsets).
  const uint64_t src =
      (uint64_t)(uintptr_t)(x + ((size_t)b * L + t) * DMODEL) + (uint64_t)lane * 16u;
  async_row_load(lds_lane, src);
  asm volatile("s_wait_asynccnt 0x0" ::: "memory");

  const int Wmax = (L < SPAN_W) ? L : SPAN_W;
  const size_t out_b = (size_t)b * (size_t)S;

#pragma unroll
  for (int w = 1; w <= SPAN_W; ++w) {
    if (w > Wmax) break;
    // offset of width-w span group: sum_{j<w} (L - j + 1)
    const int off_w = (w - 1) * (L + 1) - ((w - 1) * w) / 2;

    // Spans starting at t (this row is the START segment).
    if (t + w <= L) {
      const int s = off_w + t;
      float* dst = out + (out_b + (size_t)s) * ROW_F;  // segment 0
      async_row_store((uint64_t)(uintptr_t)dst + (uint64_t)lane * 16u, lds_lane);
      // Width-bucket embedding, written exactly once per span (keyed on start).
      // widths 1..8 -> buckets {1,2,3,4,5,5,6,7}
      if (lane < WDIM) {
        const int bucket = (w <= 5) ? w : (w - 1);
        __builtin_nontemporal_store(swe[bucket * WDIM + lane],
                                    dst + 2 * DMODEL + lane);
      }
    }

    // Spans ending at t (this row is the END segment): start = t - w + 1 >= 0.
    if (w <= t + 1) {
      const int s = off_w + (t - w + 1);
      float* dst = out + (out_b + (size_t)s) * ROW_F + DMODEL;  // segment 1
      async_row_store((uint64_t)(uintptr_t)dst + (uint64_t)lane * 16u, lds_lane);
    }
  }
  // s_endpgm performs an implicit wait-idle, draining remaining async stores.
}

extern "C" void kernel_launch(void* const* d_in, const int* in_sizes, int n_in,
                              void* d_out, int out_size, void* d_ws, size_t ws_size,
                              hipStream_t stream) {
  (void)n_in;
  (void)out_size;
  (void)d_ws;
  (void)ws_size;

  const float* x = (const float*)d_in[0];    // (8, L, 768) f32
  const float* swe = (const float*)d_in[1];  // (14, 20) f32
  float* out = (float*)d_out;

  const int B = 8;
  const int L = in_sizes[0] / (B * DMODEL);  // 512 for this harness
  const int Wmax = (L < SPAN_W) ? L : SPAN_W;
  int S = 0;
  for (int w = 1; w <= Wmax; ++w) S += L - w + 1;  // 4068

  dim3 grid((unsigned)L, (unsigned)B, 1);
  SpanRepresentation_kernel<<<grid, 32, 0, stream>>>(x, swe, out, L, S);
}